// BatchAllTripletLoss_2087354106622
// MI455X (gfx1250) — compile-verified
//
#include <hip/hip_runtime.h>
#include <hip/hip_bf16.h>

typedef __attribute__((ext_vector_type(2))) float v2f;
typedef __attribute__((ext_vector_type(8))) float v8f;

// ---------------------------------------------------------------------------
// Kernel 1: G = E * E^T via V_WMMA_F32_16X16X4_F32, one wave per 16x16 tile.
// A (16x4 f32): lanes 0-15 hold M=0..15 with K = k,k+1 in v[0],v[1];
//               lanes 16-31 hold M=0..15 with K = k+2,k+3.  (ISA 7.12.2)
// B (4x16 f32): mirrored row-striped layout; B[k,n] = E[n,k] so each lane
//               loads the same 2 consecutive floats from row (tj*16 + lane%16).
// C/D (16x16 f32): VGPR r -> row r (lanes 0-15) / row r+8 (lanes 16-31).
// ---------------------------------------------------------------------------
__global__ __launch_bounds__(32) void gram_wmma_kernel(const float* __restrict__ E,
                                                       float* __restrict__ G,
                                                       int B, int D) {
  const int nTilesX = B >> 4;
  const int ti   = blockIdx.x / nTilesX;
  const int tj   = blockIdx.x % nTilesX;
  const int lane = threadIdx.x & 31;
  const int half = lane >> 4;   // 0: K pair {0,1}, 1: K pair {2,3}
  const int l15  = lane & 15;

  const float* __restrict__ arow = E + (size_t)(ti * 16 + l15) * D + 2 * half;
  const float* __restrict__ brow = E + (size_t)(tj * 16 + l15) * D + 2 * half;

  v8f c = {};
#pragma unroll 4
  for (int k = 0; k < D; k += 4) {
    v2f a = *(const v2f*)(arow + k);   // global_load_b64
    v2f b = *(const v2f*)(brow + k);   // global_load_b64
    // 8 args: (neg_a, A, neg_b, B, c_mod, C, reuse_a, reuse_b)
    c = __builtin_amdgcn_wmma_f32_16x16x4_f32(false, a, false, b,
                                              (short)0, c, false, false);
  }

  const int row0 = ti * 16 + 8 * half;
  const int col  = tj * 16 + l15;
#pragma unroll
  for (int r = 0; r < 8; ++r) {
    G[(size_t)(row0 + r) * B + col] = c[r];
  }
}

// ---------------------------------------------------------------------------
// Kernel 2: pairwise Euclidean distance from the Gram matrix, reference math:
//   v = max(g_ii - 2 g_ij + g_jj, 0);  d = (v==0) ? 0 : sqrt(v) + eps
// ---------------------------------------------------------------------------
__global__ __launch_bounds__(256) void dist_kernel(const float* __restrict__ G,
                                                   float* __restrict__ Dm, int B) {
  const int idx = blockIdx.x * blockDim.x + threadIdx.x;
  if (idx >= B * B) return;
  const int i = idx / B;
  const int j = idx - i * B;
  const float sqi = G[(size_t)i * B + i];
  const float sqj = G[(size_t)j * B + j];
  float v = sqi - 2.0f * G[idx] + sqj;
  v = fmaxf(v, 0.0f);
  Dm[idx] = (v == 0.0f) ? 0.0f : (__builtin_sqrtf(v) + 1e-5f);
}

// ---------------------------------------------------------------------------
// Kernel 3: batch-all triplet reduction, one block (8 waves) per anchor `a`.
// mask(a,p,n) collapses to: p != a, lab[p]==lab[a], lab[n]!=lab[a]
// (those conditions already imply n!=a and n!=p).
// num_valid contribution per anchor = npos * nneg (computed analytically).
// Per-anchor partials -> workspace (deterministic; no float atomics).
// ---------------------------------------------------------------------------
__global__ __launch_bounds__(256) void triplet_kernel(const float* __restrict__ Dm,
                                                      const long long* __restrict__ labels,
                                                      float* __restrict__ partials,
                                                      int B) {
  __shared__ float drow[512];
  __shared__ int   lab[512];
  __shared__ float redS[256];
  __shared__ int   redC[256];
  __shared__ int   redP[256];
  __shared__ int   redN[256];

  const int a = blockIdx.x;
  const int t = threadIdx.x;

  for (int j = t; j < B; j += 256) {
    drow[j] = Dm[(size_t)a * B + j];
    lab[j]  = (int)labels[j];
  }
  __syncthreads();

  const int la = lab[a];

  int npos = 0, nneg = 0;
  for (int j = t; j < B; j += 256) {
    const int lj = lab[j];
    if (j != a && lj == la) ++npos;
    if (lj != la) ++nneg;
  }

  float sum = 0.0f;
  int   cpos = 0;
  for (int p = 0; p < B; ++p) {
    if (p == a || lab[p] != la) continue;
    const float dp = drow[p];
    for (int n = t; n < B; n += 256) {
      if (lab[n] != la) {
        const float diff = dp - drow[n];
        if (diff > 0.0f)    sum += diff;
        if (diff > 1e-5f)   ++cpos;
      }
    }
  }

  redS[t] = sum; redC[t] = cpos; redP[t] = npos; redN[t] = nneg;
  __syncthreads();
#pragma unroll
  for (int s = 128; s > 0; s >>= 1) {
    if (t < s) {
      redS[t] += redS[t + s];
      redC[t] += redC[t + s];
      redP[t] += redP[t + s];
      redN[t] += redN[t + s];
    }
    __syncthreads();
  }

  if (t == 0) {
    partials[a]         = redS[0];                              // sum of tl
    partials[B + a]     = (float)redC[0];                       // num_positive
    partials[2 * B + a] = (float)redP[0] * (float)redN[0];      // num_valid
  }
}

// ---------------------------------------------------------------------------
// Kernel 4: deterministic final reduction over 512 per-anchor partials.
// ---------------------------------------------------------------------------
__global__ __launch_bounds__(512) void final_kernel(const float* __restrict__ partials,
                                                    float* __restrict__ out, int B) {
  __shared__ float s0[512];
  __shared__ float s1[512];
  __shared__ float s2[512];
  const int t = threadIdx.x;
  s0[t] = (t < B) ? partials[t]         : 0.0f;
  s1[t] = (t < B) ? partials[B + t]     : 0.0f;
  s2[t] = (t < B) ? partials[2 * B + t] : 0.0f;
  __syncthreads();
#pragma unroll
  for (int s = 256; s > 0; s >>= 1) {
    if (t < s) { s0[t] += s0[t + s]; s1[t] += s1[t + s]; s2[t] += s2[t + s]; }
    __syncthreads();
  }
  if (t == 0) {
    out[0] = s0[0] / (s1[0] + 1e-5f);   // loss
    out[1] = s1[0] / (s2[0] + 1e-5f);   // fraction_positive
  }
}

extern "C" void kernel_launch(void* const* d_in, const int* in_sizes, int n_in,
                              void* d_out, int out_size, void* d_ws, size_t ws_size,
                              hipStream_t stream) {
  const float*     E      = (const float*)d_in[0];
  const long long* labels = (const long long*)d_in[1];   // int64 labels
  float*           out    = (float*)d_out;

  const int B = in_sizes[1];             // 512
  const int D = in_sizes[0] / B;         // 512

  float* G        = (float*)d_ws;                 // B*B floats
  float* Dm       = G + (size_t)B * B;            // B*B floats
  float* partials = Dm + (size_t)B * B;           // 3*B floats

  const int nTiles = (B / 16) * (B / 16);         // 1024 waves
  gram_wmma_kernel<<<nTiles, 32, 0, stream>>>(E, G, B, D);

  dist_kernel<<<(B * B + 255) / 256, 256, 0, stream>>>(G, Dm, B);

  triplet_kernel<<<B, 256, 0, stream>>>(Dm, labels, partials, B);

  final_kernel<<<1, 512, 0, stream>>>(partials, out, B);
}